// GumbelSampler_49598282334661
// MI455X (gfx1250) — compile-verified
//
#include <hip/hip_runtime.h>
#include <math.h>

// ---------------------------------------------------------------------------
// Gumbel top-K edge sampler for MI455X (gfx1250, wave32).
// One workgroup (1024 threads = 32 waves) per (batch, ensemble) row.
// fs / khot state streamed through the block's own strided slice of d_out
// (exactly 2*131072 slots per block) -> fully L2-resident on the 192MB L2,
// no d_ws dependence, no inter-block races.
// Row-wide SUM reductions use V_WMMA_F32_16X16X4_F32 (exact f32 math).
// fs/TAU computed as fs*10.0f (monotonic, exact constant) to avoid the
// ~10-op IEEE divide sequence per element that hipcc emits for '/ 0.1f'.
// ---------------------------------------------------------------------------

typedef float v2f __attribute__((ext_vector_type(2)));
typedef float v8f __attribute__((ext_vector_type(8)));

#define NMAXD 512
#define ENSD  4
#define NTRIU 130816      // 512*511/2
#define NPAD  131072      // 1024 * 128
#define KSEL  16
#define NTHR  1024
#define EPT   128         // elements per thread
#define INVTAU 10.0f      // 1/TAU, exact in f32
#define EPSF  1.17549435082228750797e-38f   // np.float32 tiny

__device__ __forceinline__ int triu_start(int i) {
  // row i of strict upper triangle starts at i*(2N-1-i)/2, N=512
  return (i * (1023 - i)) >> 1;
}

__device__ __forceinline__ void triu_decode(int t, int& oi, int& oj) {
  int i = (int)((1023.0f - sqrtf(1046529.0f - 8.0f * (float)t)) * 0.5f);
  if (i < 0) i = 0;
  if (i > 510) i = 510;
  while (triu_start(i + 1) <= t) ++i;
  while (triu_start(i) > t) --i;
  oi = i;
  oj = t - triu_start(i) + i + 1;
}

// Exact-f32 sum of all 32 lanes using the matrix pipe:
// A = (x in K0/K2 columns, 0 in K1/K3), B = ones  =>  D[m,n] = x[m] + x[m+16].
// Summing D's 8 VGPRs gives half-sums; one xor-16 shuffle completes it.
__device__ __forceinline__ float wave_sum32(float x) {
  v2f a; a[0] = x;    a[1] = 0.0f;
  v2f bo; bo[0] = 1.0f; bo[1] = 1.0f;
  v8f c = {0.0f, 0.0f, 0.0f, 0.0f, 0.0f, 0.0f, 0.0f, 0.0f};
  v8f d = __builtin_amdgcn_wmma_f32_16x16x4_f32(false, a, false, bo,
                                                (short)0, c, false, false);
  float s = ((d[0] + d[1]) + (d[2] + d[3])) + ((d[4] + d[5]) + (d[6] + d[7]));
  s += __shfl_xor(s, 16, 32);
  return s;
}

__device__ __forceinline__ float wave_max32(float v) {
#pragma unroll
  for (int off = 16; off >= 1; off >>= 1)
    v = fmaxf(v, __shfl_xor(v, off, 32));
  return v;
}

__device__ __forceinline__ float block_sum(float v, float* sred) {
  const int lane = threadIdx.x & 31, wave = threadIdx.x >> 5;
  v = wave_sum32(v);
  __syncthreads();
  if (lane == 0) sred[wave] = v;
  __syncthreads();
  return wave_sum32(sred[lane]);   // every wave computes the same total
}

__device__ __forceinline__ float block_max(float v, float* sred) {
  const int lane = threadIdx.x & 31, wave = threadIdx.x >> 5;
  v = wave_max32(v);
  __syncthreads();
  if (lane == 0) sred[wave] = v;
  __syncthreads();
  return wave_max32(sred[lane]);
}

__device__ __forceinline__ void block_argmax(float& v, int& idx,
                                             float* sv, int* si) {
#pragma unroll
  for (int off = 16; off >= 1; off >>= 1) {
    float ov = __shfl_xor(v, off, 32);
    int   oi = __shfl_xor(idx, off, 32);
    if (ov > v || (ov == v && oi < idx)) { v = ov; idx = oi; }
  }
  const int lane = threadIdx.x & 31, wave = threadIdx.x >> 5;
  __syncthreads();
  if (lane == 0) { sv[wave] = v; si[wave] = idx; }
  __syncthreads();
  v = sv[lane]; idx = si[lane];
#pragma unroll
  for (int off = 16; off >= 1; off >>= 1) {
    float ov = __shfl_xor(v, off, 32);
    int   oi = __shfl_xor(idx, off, 32);
    if (ov > v || (ov == v && oi < idx)) { v = ov; idx = oi; }
  }
}

__global__ void __launch_bounds__(NTHR)
gumbel_topk_kernel(const float* __restrict__ scores,
                   const float* __restrict__ g,
                   float* __restrict__ out) {
  __shared__ float sred[32];
  __shared__ float sv[32];
  __shared__ int   si[32];
  __shared__ int   wt[KSEL];
  __shared__ float wv[KSEL];

  const int tid = threadIdx.x;
  const int row = blockIdx.x;          // row = b*ENS + e
  const int b = row >> 2, e = row & 3;
  const int obase = b * (NMAXD * NMAXD * ENSD) + e;
  // FS slot t  -> out[obase + t*4];  KH slot t -> out[obase + (NPAD+t)*4]
  // (block-private: this is exactly the block's own strided output slice)

  float Mx = 0.0f, invS = 1.0f;        // scaled max and 1/sum of prev softmax

  for (int it = 0; it < KSEL; ++it) {
    float lmax = -__builtin_inff();    // tracks max of fs (unscaled)
    if (it == 0) {
      // gather + init pass: fs = s[b,i,j,e] + s[b,j,i,e] + gumbel;  kh = 0
      for (int k = 0; k < EPT; ++k) {
        int t = k * NTHR + tid;
        float fs;
        if (t < NTRIU) {
          int i, j; triu_decode(t, i, j);
          fs = scores[((b * NMAXD + i) * NMAXD + j) * ENSD + e]
             + scores[((b * NMAXD + j) * NMAXD + i) * ENSD + e]
             + g[row * NTRIU + t];
        } else {
          fs = -__builtin_inff();      // padding: exp()=0, never selected
        }
        out[obase + t * ENSD] = fs;
        out[obase + (NPAD + t) * ENSD] = 0.0f;
        lmax = fmaxf(lmax, fs);
      }
    } else {
      // fold previous iteration's onehot: kh += oh; fs += log(max(1-oh,eps))
      for (int k = 0; k < EPT; ++k) {
        int t = k * NTHR + tid;
        float fs = out[obase + t * ENSD];
        float oh = expf(fs * INVTAU - Mx) * invS;
        out[obase + (NPAD + t) * ENSD] += oh;
        fs += logf(fmaxf(1.0f - oh, EPSF));
        out[obase + t * ENSD] = fs;
        lmax = fmaxf(lmax, fs);
      }
    }
    // max commutes with the monotonic rounded multiply by +10.0f
    float M = block_max(lmax, sred) * INVTAU;
    float lsum = 0.0f;
    for (int k = 0; k < EPT; ++k) {
      int t = k * NTHR + tid;
      lsum += expf(out[obase + t * ENSD] * INVTAU - M);
    }
    float S = block_sum(lsum, sred);   // WMMA-based exact f32 reduction
    Mx = M;
    invS = 1.0f / S;
  }

  // accumulate the final iteration's onehot into khot
  for (int k = 0; k < EPT; ++k) {
    int t = k * NTHR + tid;
    float fs = out[obase + t * ENSD];
    out[obase + (NPAD + t) * ENSD] += expf(fs * INVTAU - Mx) * invS;
  }

  // top-16 of khot: 16 argmax rounds; winner slot poisoned with -inf
  for (int r = 0; r < KSEL; ++r) {
    float bv = -__builtin_inff();
    int   bt = 0x7fffffff;
    for (int k = 0; k < EPT; ++k) {
      int t = k * NTHR + tid;
      if (t >= NTRIU) continue;
      float kv = out[obase + (NPAD + t) * ENSD];
      if (kv > bv || (kv == bv && t < bt)) { bv = kv; bt = t; }
    }
    block_argmax(bv, bt, sv, si);                 // uniform winner
    if ((bt & (NTHR - 1)) == tid)                 // owner thread marks it
      out[obase + (NPAD + bt) * ENSD] = -__builtin_inff();
    if (tid == 0) { wt[r] = bt; wv[r] = bv; }
  }
  __syncthreads();

  // zero this block's entire output slice (which doubled as scratch)
  for (int s = tid; s < 2 * NPAD; s += NTHR)
    out[obase + s * ENSD] = 0.0f;
  asm volatile("s_wait_storecnt 0x0" ::: "memory");  // zeros reach L2 first
  __syncthreads();

  // scatter res = (1 + khot) - khot symmetrically at the 16 winners
  if (tid == 0) {
#pragma unroll
    for (int r = 0; r < KSEL; ++r) {
      int t = wt[r];
      float kv = wv[r];
      float res = (1.0f + kv) - kv;
      int i, j; triu_decode(t, i, j);
      out[((b * NMAXD + i) * NMAXD + j) * ENSD + e] = res;
      out[((b * NMAXD + j) * NMAXD + i) * ENSD + e] = res;
    }
  }
}

extern "C" void kernel_launch(void* const* d_in, const int* in_sizes, int n_in,
                              void* d_out, int out_size, void* d_ws,
                              size_t ws_size, hipStream_t stream) {
  (void)in_sizes; (void)n_in; (void)d_ws; (void)ws_size; (void)out_size;
  const float* scores = (const float*)d_in[0];
  const float* g      = (const float*)d_in[1];
  float* out          = (float*)d_out;
  hipLaunchKernelGGL(gumbel_topk_kernel, dim3(32), dim3(NTHR), 0, stream,
                     scores, g, out);
}